// SUNConv_38293928411681
// MI455X (gfx1250) — compile-verified
//
#include <hip/hip_runtime.h>
#include <hip/hip_bf16.h>

// ---------------------------------------------------------------------------
// SUNConv for MI455X (gfx1250).
// Sizes fixed by the reference.
// ---------------------------------------------------------------------------
static constexpr int   N  = 1024;     // nodes
static constexpr int   M  = 131072;   // tuples
static constexpr int   E  = 32768;    // edges
static constexpr int   D  = 64;       // embedding dim

typedef __attribute__((ext_vector_type(2))) float v2f;
typedef __attribute__((ext_vector_type(8))) float v8f;

// ---------------------------------------------------------------------------
// Workspace layout (float-index offsets). Total ~337 MB.
// Zero region = [Xd1 | diag | p | q | x5 | colcnt | colfill]  (one memset)
// ---------------------------------------------------------------------------
static constexpr size_t OFF_XD1     = 0;                          // N*N*D
static constexpr size_t OFF_DIAG    = OFF_XD1   + (size_t)N*N*D;  // N*D
static constexpr size_t OFF_P       = OFF_DIAG  + (size_t)N*D;    // N*D
static constexpr size_t OFF_Q       = OFF_P     + (size_t)N*D;    // N*D
static constexpr size_t OFF_X5      = OFF_Q     + (size_t)N*D;    // N*D
static constexpr size_t OFF_COLCNT  = OFF_X5    + (size_t)N*D;    // N ints
static constexpr size_t OFF_COLFILL = OFF_COLCNT + N;             // N ints
static constexpr size_t ZERO_FLOATS = OFF_COLFILL + N;            // end of zero region
static constexpr size_t OFF_COLSTART= ZERO_FLOATS;                // N+1 ints (pad 1028)
static constexpr size_t OFF_CSCROW  = OFF_COLSTART + 1028;        // E ints
static constexpr size_t OFF_CSCVAL  = OFF_CSCROW  + E;            // E floats
static constexpr size_t OFF_XV1     = OFF_CSCVAL  + E;            // M*D
static constexpr size_t OFF_X1      = OFF_XV1     + (size_t)M*D;  // M*D
static constexpr size_t OFF_UI      = OFF_X1      + (size_t)M*D;  // N*D
static constexpr size_t OFF_VJ      = OFF_UI      + (size_t)N*D;  // N*D

// ---------------------------------------------------------------------------
// CSC build for the adjacency A (column = aj, row = ai, value = Av).
// Duplicate (ai,aj) pairs stay as separate entries; the gather sums them,
// matching the reference's scatter-add into dense Ad.
// ---------------------------------------------------------------------------
__global__ void k_colcount(const int* __restrict__ aj, int* __restrict__ colcnt) {
    int e = blockIdx.x * blockDim.x + threadIdx.x;
    if (e < E) atomicAdd(&colcnt[aj[e]], 1);
}

__global__ __launch_bounds__(1024) void k_scan(const int* __restrict__ colcnt,
                                               int* __restrict__ colstart) {
    __shared__ int s[N];
    int t = threadIdx.x;
    s[t] = colcnt[t];
    __syncthreads();
    #pragma unroll
    for (int off = 1; off < N; off <<= 1) {
        int v = (t >= off) ? s[t - off] : 0;
        __syncthreads();
        s[t] += v;
        __syncthreads();
    }
    colstart[t + 1] = s[t];           // inclusive -> exclusive shift
    if (t == 0) colstart[0] = 0;
}

__global__ void k_cscfill(const int* __restrict__ ai, const int* __restrict__ aj,
                          const float* __restrict__ Av,
                          const int* __restrict__ colstart, int* __restrict__ colfill,
                          int* __restrict__ cscrow, float* __restrict__ cscval) {
    int e = blockIdx.x * blockDim.x + threadIdx.x;
    if (e >= E) return;
    int c = aj[e];
    int pos = colstart[c] + atomicAdd(&colfill[c], 1);
    cscrow[pos] = ai[e];
    cscval[pos] = Av[e];
}

// ---------------------------------------------------------------------------
// Shared WMMA core:  C(16x64 strip) = A(16x64) @ W(64x64) with fp32 WMMA.
//
// 16x16x4 fp32 fragment layout (ISA 7.12.2):
//   A (16x4): lane%16 = row M; VGPR0/1 hold K = khalf / khalf+1,
//             khalf = (lane>=16 ? 2 : 0)
//   B (4x16): lane%16 = col N; VGPR0/1 hold K rows khalf / khalf+1
//   C (16x16): VGPR v -> row v (lanes 0-15) or v+8 (lanes 16-31), col = lane%16
//
// B fragments {W[k][c], W[k+1][c]} sit 256 B apart in row-major W, which
// forces ds_load_2addr + register-pack movs in the WMMA stream. So we stage
// W into LDS *fragment-packed*: Wf[kp*64 + c] = float2{W[2kp][c], W[2kp+1][c]}
// (kp = 0..31). Every B fragment is then ONE aligned ds_load_b64 directly
// into an even-aligned VGPR pair — no shuffles between the WMMAs.
// ---------------------------------------------------------------------------
__device__ __forceinline__ void stage_w_packed(const float* __restrict__ W,
                                               float2* __restrict__ Wf,
                                               int tid /*0..255*/) {
    for (int idx = tid; idx < 32 * D; idx += 256) {   // kp in [0,32), c in [0,64)
        int kp = idx >> 6, c = idx & 63;
        float2 f;
        f.x = W[(2 * kp)     * D + c];
        f.y = W[(2 * kp + 1) * D + c];
        Wf[idx] = f;
    }
}

__device__ __forceinline__ void wmma_strip_16x64(const float* __restrict__ arow,
                                                 const float2* __restrict__ Wf,
                                                 int lane, v8f acc[4]) {
    const int col = lane & 15;
    const int hi  = lane >> 4;              // 0 or 1
    #pragma unroll
    for (int kb = 0; kb < 16; ++kb) {
        const int kbase = kb * 4 + hi * 2;  // K start for this lane's fragment
        const int kp    = kb * 2 + hi;      // packed-pair index
        v2f a;
        { const float2 t = *(const float2*)(arow + kbase); a.x = t.x; a.y = t.y; }
        #pragma unroll
        for (int n = 0; n < 4; ++n) {
            const float2 bf = Wf[kp * D + n * 16 + col];   // one ds_load_b64
            v2f b; b.x = bf.x; b.y = bf.y;
            acc[n] = __builtin_amdgcn_wmma_f32_16x16x4_f32(
                false, a, false, b, (short)0, acc[n], false, false);
        }
    }
}

// ---------------------------------------------------------------------------
// GEMM1:  Xv1 = relu(Xv @ W1 + b1)   (M x 64) @ (64 x 64), fp32 WMMA.
// One wave computes a 16x64 row strip = four 16x16 C tiles.
// ---------------------------------------------------------------------------
__global__ __launch_bounds__(256) void k_gemm1(const float* __restrict__ Xv,
                                               const float* __restrict__ W1,
                                               const float* __restrict__ b1,
                                               float* __restrict__ Xv1) {
    __shared__ float2 Wf[32 * D];     // 16 KB, fragment-packed W1
    stage_w_packed(W1, Wf, threadIdx.x);
    __syncthreads();

    const int wave = threadIdx.x >> 5;
    const int lane = threadIdx.x & 31;
    const size_t m0 = ((size_t)blockIdx.x * 8 + wave) * 16;
    const int col   = lane & 15;
    const float* arow = Xv + (m0 + col) * D;   // A row index == lane%16

    v8f acc[4] = {};
    wmma_strip_16x64(arow, Wf, lane, acc);

    const int rbase = (lane >> 4) * 8;
    #pragma unroll
    for (int n = 0; n < 4; ++n) {
        const int colg = n * 16 + col;
        const float bias = b1[colg];
        #pragma unroll
        for (int v = 0; v < 8; ++v) {
            float x = acc[n][v] + bias;
            Xv1[(m0 + rbase + v) * D + colg] = x > 0.f ? x : 0.f;
        }
    }
}

// ---------------------------------------------------------------------------
// Scatter: Xd1[ti,tj,:] += Xv1 ; p[ti] += Xv ; q[tj] += Xv ; diag += Xv on ti==tj
// ---------------------------------------------------------------------------
__global__ void k_scatter(const float* __restrict__ Xv, const float* __restrict__ Xv1,
                          const int* __restrict__ ti, const int* __restrict__ tj,
                          float* __restrict__ Xd1, float* __restrict__ diag,
                          float* __restrict__ p, float* __restrict__ q) {
    int idx = blockIdx.x * blockDim.x + threadIdx.x;   // M * 16 threads (quad of d)
    if (idx >= M * (D / 4)) return;
    int m  = idx >> 4;
    int d0 = (idx & 15) * 4;
    int i = ti[m], j = tj[m];
    float4 x1v = *(const float4*)(Xv1 + (size_t)m * D + d0);
    float4 xv  = *(const float4*)(Xv  + (size_t)m * D + d0);

    float* xd = Xd1 + ((size_t)i * N + j) * D + d0;
    atomicAdd(xd + 0, x1v.x); atomicAdd(xd + 1, x1v.y);
    atomicAdd(xd + 2, x1v.z); atomicAdd(xd + 3, x1v.w);

    float* pp = p + (size_t)i * D + d0;
    atomicAdd(pp + 0, xv.x); atomicAdd(pp + 1, xv.y);
    atomicAdd(pp + 2, xv.z); atomicAdd(pp + 3, xv.w);

    float* qq = q + (size_t)j * D + d0;
    atomicAdd(qq + 0, xv.x); atomicAdd(qq + 1, xv.y);
    atomicAdd(qq + 2, xv.z); atomicAdd(qq + 3, xv.w);

    if (i == j) {
        float* dg = diag + (size_t)i * D + d0;
        atomicAdd(dg + 0, xv.x); atomicAdd(dg + 1, xv.y);
        atomicAdd(dg + 2, xv.z); atomicAdd(dg + 3, xv.w);
    }
}

// ---------------------------------------------------------------------------
// x5 = Ad @ q  done edge-wise: x5[ai_e] += Av_e * q[aj_e]
// ---------------------------------------------------------------------------
__global__ void k_x5(const int* __restrict__ ai, const int* __restrict__ aj,
                     const float* __restrict__ Av,
                     const float* __restrict__ q, float* __restrict__ x5) {
    int idx = blockIdx.x * blockDim.x + threadIdx.x;   // E * 16
    if (idx >= E * (D / 4)) return;
    int e  = idx >> 4;
    int d0 = (idx & 15) * 4;
    int r = ai[e], c = aj[e];
    float a = Av[e];
    float4 qq = *(const float4*)(q + (size_t)c * D + d0);
    float* x = x5 + (size_t)r * D + d0;
    atomicAdd(x + 0, a * qq.x); atomicAdd(x + 1, a * qq.y);
    atomicAdd(x + 2, a * qq.z); atomicAdd(x + 3, a * qq.w);
}

// ---------------------------------------------------------------------------
// Node-side W2 projections:
//   Ui = diag@W2[64:128] + q@W2[256:320] + x5@W2[320:384] + b2   (added at ti)
//   Vj = diag@W2[128:192] + p@W2[192:256]                         (added at tj)
// ---------------------------------------------------------------------------
__global__ void k_nodeproj(const float* __restrict__ diag, const float* __restrict__ p,
                           const float* __restrict__ q, const float* __restrict__ x5,
                           const float* __restrict__ W2, const float* __restrict__ b2,
                           float* __restrict__ Ui, float* __restrict__ Vj) {
    int idx = blockIdx.x * blockDim.x + threadIdx.x;   // N * D
    if (idx >= N * D) return;
    int n = idx >> 6, c = idx & 63;
    float ui = b2[c], vj = 0.f;
    #pragma unroll 4
    for (int k = 0; k < D; ++k) {
        float dg = diag[(size_t)n * D + k];
        float qv = q [(size_t)n * D + k];
        float x5v= x5[(size_t)n * D + k];
        float pv = p [(size_t)n * D + k];
        ui = fmaf(dg,  W2[( 64 + k) * D + c], ui);
        ui = fmaf(qv,  W2[(256 + k) * D + c], ui);
        ui = fmaf(x5v, W2[(320 + k) * D + c], ui);
        vj = fmaf(dg,  W2[(128 + k) * D + c], vj);
        vj = fmaf(pv,  W2[(192 + k) * D + c], vj);
    }
    Ui[idx] = ui;
    Vj[idx] = vj;
}

// ---------------------------------------------------------------------------
// Sparse x1 gather (the roofline-critical kernel, ~1 GB of Xd1 reads):
//   x1[m,:] = sum over CSC column tj[m] of  Av_e * Xd1[ti[m], ai_e, :]
// One wave per tuple; each lane owns a float2 of the 64-wide feature row,
// so each adjacency row is one coalesced 256 B read.
// ---------------------------------------------------------------------------
__global__ __launch_bounds__(256) void k_x1gather(const int* __restrict__ ti,
                                                  const int* __restrict__ tj,
                                                  const int* __restrict__ colstart,
                                                  const int* __restrict__ cscrow,
                                                  const float* __restrict__ cscval,
                                                  const float* __restrict__ Xd1,
                                                  float* __restrict__ x1) {
    int wid  = (blockIdx.x * blockDim.x + threadIdx.x) >> 5;   // tuple id
    int lane = threadIdx.x & 31;
    if (wid >= M) return;
    int i = ti[wid], c = tj[wid];
    int e0 = colstart[c], e1 = colstart[c + 1];
    const float* base = Xd1 + (size_t)i * (size_t)(N * D);
    float ax = 0.f, ay = 0.f;
    for (int e = e0; e < e1; ++e) {
        int   j = cscrow[e];
        float v = cscval[e];
        if (e + 1 < e1)
            __builtin_prefetch(base + (size_t)cscrow[e + 1] * D + lane * 2, 0, 3);
        float2 xr = *(const float2*)(base + (size_t)j * D + lane * 2);
        ax = fmaf(v, xr.x, ax);
        ay = fmaf(v, xr.y, ay);
    }
    float2 out; out.x = ax; out.y = ay;
    *(float2*)(x1 + (size_t)wid * D + lane * 2) = out;
}

// ---------------------------------------------------------------------------
// GEMM2 + fused epilogue:
//   out = relu( x1 @ W2[0:64]  + Ui[ti]  + Vj[tj] )     (b2 folded into Ui)
// Same fp32 WMMA tiling as GEMM1; the epilogue gathers Ui/Vj per output row.
// ---------------------------------------------------------------------------
__global__ __launch_bounds__(256) void k_gemm2(const float* __restrict__ x1,
                                               const float* __restrict__ W2,
                                               const int* __restrict__ ti,
                                               const int* __restrict__ tj,
                                               const float* __restrict__ Ui,
                                               const float* __restrict__ Vj,
                                               float* __restrict__ out) {
    __shared__ float2 Wf[32 * D];     // W2 rows 0..63 (the x1 block), packed
    stage_w_packed(W2, Wf, threadIdx.x);
    __syncthreads();

    const int wave = threadIdx.x >> 5;
    const int lane = threadIdx.x & 31;
    const size_t m0 = ((size_t)blockIdx.x * 8 + wave) * 16;
    const int col   = lane & 15;
    const float* arow = x1 + (m0 + col) * D;

    v8f acc[4] = {};
    wmma_strip_16x64(arow, Wf, lane, acc);

    const int rbase = (lane >> 4) * 8;
    #pragma unroll
    for (int v = 0; v < 8; ++v) {
        const size_t r = m0 + rbase + v;
        const int tiv = ti[r], tjv = tj[r];
        #pragma unroll
        for (int n = 0; n < 4; ++n) {
            const int colg = n * 16 + col;
            float x = acc[n][v] + Ui[(size_t)tiv * D + colg] + Vj[(size_t)tjv * D + colg];
            out[r * D + colg] = x > 0.f ? x : 0.f;
        }
    }
}

// ---------------------------------------------------------------------------
// Host-side orchestration (graph-capture safe: only async memset + launches).
// ---------------------------------------------------------------------------
extern "C" void kernel_launch(void* const* d_in, const int* in_sizes, int n_in,
                              void* d_out, int out_size, void* d_ws, size_t ws_size,
                              hipStream_t stream) {
    const float* Xv = (const float*)d_in[0];
    const float* Av = (const float*)d_in[1];
    const float* W1 = (const float*)d_in[2];
    const float* b1 = (const float*)d_in[3];
    const float* W2 = (const float*)d_in[4];
    const float* b2 = (const float*)d_in[5];
    const int*   ti = (const int*)d_in[6];          // tind[0]
    const int*   tj = (const int*)d_in[6] + M;      // tind[1]
    const int*   ai = (const int*)d_in[7];          // aind[0]
    const int*   aj = (const int*)d_in[7] + E;      // aind[1]
    float*       out = (float*)d_out;

    float* ws = (float*)d_ws;
    float* Xd1     = ws + OFF_XD1;
    float* diag    = ws + OFF_DIAG;
    float* p       = ws + OFF_P;
    float* q       = ws + OFF_Q;
    float* x5      = ws + OFF_X5;
    int*   colcnt  = (int*)(ws + OFF_COLCNT);
    int*   colfill = (int*)(ws + OFF_COLFILL);
    int*   colstart= (int*)(ws + OFF_COLSTART);
    int*   cscrow  = (int*)(ws + OFF_CSCROW);
    float* cscval  = ws + OFF_CSCVAL;
    float* Xv1     = ws + OFF_XV1;
    float* x1      = ws + OFF_X1;
    float* Ui      = ws + OFF_UI;
    float* Vj      = ws + OFF_VJ;

    // One memset for everything accumulated into (Xd1/diag/p/q/x5/counters).
    hipMemsetAsync(Xd1, 0, ZERO_FLOATS * sizeof(float), stream);

    // CSC build of adjacency.
    k_colcount<<<(E + 255) / 256, 256, 0, stream>>>(aj, colcnt);
    k_scan<<<1, 1024, 0, stream>>>(colcnt, colstart);
    k_cscfill<<<(E + 255) / 256, 256, 0, stream>>>(ai, aj, Av, colstart, colfill,
                                                   cscrow, cscval);

    // Xv1 = relu(Xv @ W1 + b1)   [fp32 WMMA]
    k_gemm1<<<M / 128, 256, 0, stream>>>(Xv, W1, b1, Xv1);

    // Densify Xv1 on the tuple pattern + node pooling sums.
    k_scatter<<<(M * (D / 4)) / 256, 256, 0, stream>>>(Xv, Xv1, ti, tj,
                                                       Xd1, diag, p, q);

    // x5 = Ad @ q (edge-wise), needs q complete (stream-ordered).
    k_x5<<<(E * (D / 4)) / 256, 256, 0, stream>>>(ai, aj, Av, q, x5);

    // Node-side W2 projections (b2 folded into Ui).
    k_nodeproj<<<(N * D) / 256, 256, 0, stream>>>(diag, p, q, x5, W2, b2, Ui, Vj);

    // Sparse message passing: x1[m] = sum_e Av_e * Xd1[ti_m, ai_e, :].
    k_x1gather<<<(M * 32) / 256, 256, 0, stream>>>(ti, tj, colstart, cscrow,
                                                   cscval, Xd1, x1);

    // out = relu(x1 @ W2a + Ui[ti] + Vj[tj])   [fp32 WMMA + fused epilogue]
    k_gemm2<<<M / 128, 256, 0, stream>>>(x1, W2, ti, tj, Ui, Vj, out);
}